// MultiHeadAttention_41257455845741
// MI455X (gfx1250) — compile-verified
//
#include <hip/hip_runtime.h>
#include <hip/hip_bf16.h>

typedef __attribute__((ext_vector_type(2))) float v2f;
typedef __attribute__((ext_vector_type(8))) float v8f;

#define WMMA_F32(a, b, c) \
    __builtin_amdgcn_wmma_f32_16x16x4_f32(false, (a), false, (b), (short)0, (c), false, false)

constexpr int BATCH = 4;
constexpr int HEADS = 16;
constexpr int NTOK  = 2049;
constexpr int DMODEL = 1024;
constexpr int HD = 64;
constexpr int MROWS = BATCH * NTOK;                        // 8196
constexpr size_t QSZ = (size_t)BATCH * HEADS * NTOK * HD;  // floats per Q/K/V buffer

static_assert(NTOK - 1 == 2048, "rope indexing assumes N-1 == 2048");

// ---------------------------------------------------------------------------
// CDNA5 async global->LDS copy (ASYNCcnt-tracked, no VGPR round trip).
// LDS byte address = low 32 bits of the generic shared pointer (flat LDS
// aperture maps addr[31:0] to the wave's LDS allocation).
// ---------------------------------------------------------------------------
__device__ __forceinline__ void async_load_b128(uint32_t lds_addr, const float* gaddr) {
    asm volatile("global_load_async_to_lds_b128 %0, %1, off"
                 :: "v"(lds_addr), "v"(gaddr) : "memory");
}
template <int N>
__device__ __forceinline__ void wait_asynccnt() {
    asm volatile("s_wait_asynccnt %0" :: "n"(N) : "memory");
}

// ---------------------------------------------------------------------------
// DPP16 butterfly reductions across the 16-lane row (matches WMMA D-layout
// halves on wave32). quad_perm 0xB1 = xor1, 0x4E = xor2, row_half_mirror
// (0x141) acts as xor4 once quads are uniform, row_mirror (0x140) as xor8.
// ---------------------------------------------------------------------------
template <int CTRL>
__device__ __forceinline__ float dpp_mov16(float x) {
    return __int_as_float(__builtin_amdgcn_update_dpp(
        0, __float_as_int(x), CTRL, 0xf, 0xf, true));
}
__device__ __forceinline__ float row_max16(float v) {
    v = fmaxf(v, dpp_mov16<0xB1>(v));
    v = fmaxf(v, dpp_mov16<0x4E>(v));
    v = fmaxf(v, dpp_mov16<0x141>(v));
    v = fmaxf(v, dpp_mov16<0x140>(v));
    return v;
}
__device__ __forceinline__ float row_sum16(float v) {
    v += dpp_mov16<0xB1>(v);
    v += dpp_mov16<0x4E>(v);
    v += dpp_mov16<0x141>(v);
    v += dpp_mov16<0x140>(v);
    return v;
}

// ---------------------------------------------------------------------------
// GEMM: out = A(MxK) @ W(NxK)^T + bias
// Double-buffered LDS tiles filled via global_load_async_to_lds_b128; tile
// t+1 is in flight while tile t feeds 8 chained v_wmma_f32_16x16x4_f32.
// EPI 0: plain row-major (MxN) store.  EPI 1: Q/K/V split, (B,H,NTOK,HD).
// ---------------------------------------------------------------------------
template <int EPI>
__global__ __launch_bounds__(256) void gemm_wmma_kernel(
    const float* __restrict__ A, const float* __restrict__ W,
    const float* __restrict__ bias, float* __restrict__ out,
    int M, int N, int K)
{
    __shared__ __align__(16) float As[2][32][36];   // stride 36: 16B-aligned, conflict-free
    __shared__ __align__(16) float Bs[2][64][36];
    constexpr uint32_t ABUF = 32 * 36 * sizeof(float);
    constexpr uint32_t BBUF = 64 * 36 * sizeof(float);

    const int tid  = threadIdx.x;
    const int m0   = blockIdx.y * 32;
    const int n0   = blockIdx.x * 64;
    const int lane = tid & 31;
    const int lr   = lane & 15;
    const int hf   = lane >> 4;
    const int wave = tid >> 5;
    const int wm   = (wave >> 2) * 16;
    const int wn   = (wave & 3) * 16;

    const int ar = tid >> 3, ac = (tid & 7) * 4;   // A: 32x32, one b128 per thread
    const int br = tid >> 2, bc = (tid & 3) * 8;   // B: 64x32, two b128 per thread
    const int am = m0 + ar;
    // clamped row: garbage only feeds output rows that are masked at store
    const float* Aload = A + (size_t)min(am, M - 1) * K + ac;
    const float* Wload = W + (size_t)(n0 + br) * K + bc;   // N is multiple of 64

    const uint32_t asA = (uint32_t)(uintptr_t)&As[0][ar][ac];
    const uint32_t asB = (uint32_t)(uintptr_t)&Bs[0][br][bc];

    auto issue = [&](int buf, int k0) {
        async_load_b128(asA + buf * ABUF, Aload + k0);
        async_load_b128(asB + buf * BBUF,      Wload + k0);
        async_load_b128(asB + buf * BBUF + 16, Wload + k0 + 4);
    };

    v8f acc = {};
    issue(0, 0);
    for (int k0 = 0; k0 < K; k0 += 32) {
        const int cur   = (k0 >> 5) & 1;
        const bool more = (k0 + 32) < K;           // uniform
        if (more) issue(cur ^ 1, k0 + 32);         // next tile in flight during compute
        if (more) wait_asynccnt<3>(); else wait_asynccnt<0>();
        __syncthreads();                           // current tile visible to all waves
        v2f af[8], bf[8];
#pragma unroll
        for (int t8 = 0; t8 < 8; ++t8) {
            af[t8] = *(const v2f*)&As[cur][wm + lr][t8 * 4 + 2 * hf];
            bf[t8] = *(const v2f*)&Bs[cur][wn + lr][t8 * 4 + 2 * hf];
        }
#pragma unroll
        for (int t8 = 0; t8 < 8; ++t8) acc = WMMA_F32(af[t8], bf[t8], acc);
        __syncthreads();                           // done reading before buffer reuse
    }

    const int n = n0 + wn + lr;
    const float bv = bias[n];
    if (EPI == 0) {
#pragma unroll
        for (int r = 0; r < 8; ++r) {
            const int m = m0 + wm + r + 8 * hf;
            if (m < M) out[(size_t)m * N + n] = acc[r] + bv;
        }
    } else {
        const int three = n >> 10;
        const int hh    = (n >> 6) & 15;
        const int d     = n & 63;
        float* base = out + (size_t)three * QSZ;
#pragma unroll
        for (int r = 0; r < 8; ++r) {
            const int m = m0 + wm + r + 8 * hf;
            if (m < M) {
                const int b = m / NTOK;
                const int t = m - b * NTOK;
                base[(((size_t)(b * HEADS + hh)) * NTOK + t) * HD + d] = acc[r] + bv;
            }
        }
    }
}

// ---------------------------------------------------------------------------
// RoPE on Q and K, (B,H,NTOK,HD) layout; token 0 (cls) untouched.
// ---------------------------------------------------------------------------
__global__ __launch_bounds__(256) void rope_kernel(
    float* __restrict__ Q, float* __restrict__ K,
    const int* __restrict__ position)
{
    const int idx = blockIdx.x * 256 + threadIdx.x;
    constexpr int TOTAL = BATCH * HEADS * (NTOK - 1) * (HD / 2);
    if (idx >= TOTAL) return;
    const int p     = idx & 31;
    const int rest  = idx >> 5;
    const int t     = rest & 2047;
    const int rest2 = rest >> 11;
    const int hh    = rest2 & 15;
    const int b     = rest2 >> 4;

    const float posf = (float)position[b * (NTOK - 1) + t];
    const float freq = powf(10000.0f, -(float)(2 * p) / (float)HD);
    const float ang  = posf * freq;
    float s, c;
    sincosf(ang, &s, &c);

    const size_t base = (((size_t)(b * HEADS + hh)) * NTOK + (t + 1)) * HD + 2 * p;
    const float qre = Q[base], qim = Q[base + 1];
    Q[base]     = qre * c - qim * s;
    Q[base + 1] = qre * s + qim * c;
    const float kre = K[base], kim = K[base + 1];
    K[base]     = kre * c - kim * s;
    K[base + 1] = kre * s + kim * c;
}

// ---------------------------------------------------------------------------
// Flash attention: grid (ceil(NTOK/128), B*H), 256 threads = 8 waves.
// Clamp-free fast path for full key tiles (pointer arithmetic only), one
// clamped+masked tail tile. DPP16 row reductions; P via per-wave LDS.
// ---------------------------------------------------------------------------
__global__ __launch_bounds__(256) void attn_kernel(
    const float* __restrict__ Q, const float* __restrict__ K,
    const float* __restrict__ V, float* __restrict__ O, int ntok)
{
    __shared__ float Pl[8][16][18];   // per-wave 16x16 P staging

    const int tid  = threadIdx.x;
    const int wave = tid >> 5;
    const int lane = tid & 31;
    const int lr   = lane & 15;
    const int hf   = lane >> 4;

    const int bh = blockIdx.y;
    const int b  = bh >> 4;
    const int hh = bh & 15;
    const size_t hbase = (size_t)bh * ntok * HD;
    const float* Qb = Q + hbase;
    const float* Kb = K + hbase;
    const float* Vb = V + hbase;
    const float* Kp = Kb + (size_t)lr * HD + 2 * hf;   // lane's key-row base
    const float* Vp = Vb + lr;                         // lane's V-column base

    const int q0 = (blockIdx.x * 8 + wave) * 16;
    const int qr = min(q0 + lr, ntok - 1);

    // Q fragment in A-layout, pre-scaled by 1/sqrt(hd)
    v2f qf[16];
    {
        const float* qrow = Qb + (size_t)qr * HD + 2 * hf;
#pragma unroll
        for (int c = 0; c < 16; ++c) {
            v2f t = *(const v2f*)(qrow + 4 * c);
            t.x *= 0.125f; t.y *= 0.125f;
            qf[c] = t;
        }
    }

    float mrow[8], lrow[8];
    v8f ov[4] = {};
#pragma unroll
    for (int r = 0; r < 8; ++r) { mrow[r] = -3.0e38f; lrow[r] = 0.0f; }

    auto process_tile = [&](int j0, bool tail) {
        // ---- preload K fragments (one clause, one wait) ----
        v2f kb[16];
        if (!tail) {
            const float* krow = Kp + (size_t)j0 * HD;
#pragma unroll
            for (int c = 0; c < 16; ++c) kb[c] = *(const v2f*)(krow + 4 * c);
            __builtin_prefetch(krow + 16 * HD, 0, 1);   // next key tile
        } else {
            const float* krow = Kb + (size_t)min(j0 + lr, ntok - 1) * HD + 2 * hf;
#pragma unroll
            for (int c = 0; c < 16; ++c) kb[c] = *(const v2f*)(krow + 4 * c);
        }

        // ---- S = (Q/sqrt(hd)) @ K^T ----
        v8f s = {};
#pragma unroll
        for (int c = 0; c < 16; ++c) s = WMMA_F32(qf[c], kb[c], s);

        // ---- preload V fragments; latency hides under softmax VALU ----
        v2f va[4][4];   // [kk/4][group]
#pragma unroll
        for (int k4 = 0; k4 < 4; ++k4) {
            if (!tail) {
                const float* vr0 = Vp + (size_t)(j0 + k4 * 4 + 2 * hf) * HD;
#pragma unroll
                for (int g = 0; g < 4; ++g) {
                    va[k4][g].x = vr0[g * 16];
                    va[k4][g].y = vr0[HD + g * 16];
                }
            } else {
                const int v0 = min(j0 + k4 * 4 + 2 * hf,     ntok - 1);
                const int v1 = min(j0 + k4 * 4 + 2 * hf + 1, ntok - 1);
                const float* vr0 = Vb + (size_t)v0 * HD + lr;
                const float* vr1 = Vb + (size_t)v1 * HD + lr;
#pragma unroll
                for (int g = 0; g < 4; ++g) {
                    va[k4][g].x = vr0[g * 16];
                    va[k4][g].y = vr1[g * 16];
                }
            }
        }

        // ---- online softmax (DPP16 row reductions, row = r + 8*hf) ----
        const bool kvalid = tail ? ((j0 + lr) < ntok) : true;
#pragma unroll
        for (int r = 0; r < 8; ++r) {
            const float sv   = kvalid ? s[r] : -3.0e38f;
            const float mx   = row_max16(sv);
            const float mnew = fmaxf(mrow[r], mx);
            const float corr = expf(mrow[r] - mnew);
            mrow[r] = mnew;
            const float p = expf(sv - mnew);
            lrow[r] = lrow[r] * corr + row_sum16(p);
            ov[0][r] *= corr; ov[1][r] *= corr; ov[2][r] *= corr; ov[3][r] *= corr;
            Pl[wave][r + 8 * hf][lr] = p;   // D-layout -> LDS
        }
        __syncthreads();   // uniform call sites -> convergent

        // ---- O += P @ V ----
#pragma unroll
        for (int k4 = 0; k4 < 4; ++k4) {
            v2f pa = *(const v2f*)&Pl[wave][lr][k4 * 4 + 2 * hf];   // A-layout read
#pragma unroll
            for (int g = 0; g < 4; ++g) ov[g] = WMMA_F32(pa, va[k4][g], ov[g]);
        }
        __syncthreads();
    };

    int j0 = 0;
    for (; j0 + 16 <= ntok; j0 += 16) process_tile(j0, false);
    if (j0 < ntok) process_tile(j0, true);

    // ---- normalize + store in (B, N, H, HD) == (B, N, D) ----
#pragma unroll
    for (int r = 0; r < 8; ++r) {
        const int q = q0 + r + 8 * hf;
        if (q < ntok) {
            const float inv = 1.0f / lrow[r];
#pragma unroll
            for (int g = 0; g < 4; ++g)
                O[(((size_t)(b * ntok + q)) * HEADS + hh) * HD + g * 16 + lr] = ov[g][r] * inv;
        }
    }
}

// ---------------------------------------------------------------------------
extern "C" void kernel_launch(void* const* d_in, const int* in_sizes, int n_in,
                              void* d_out, int out_size, void* d_ws, size_t ws_size,
                              hipStream_t stream)
{
    const float* x        = (const float*)d_in[0];
    const int*   position = (const int*)d_in[1];
    const float* w_qkv    = (const float*)d_in[2];
    const float* b_qkv    = (const float*)d_in[3];
    const float* w_fc     = (const float*)d_in[4];
    const float* b_fc     = (const float*)d_in[5];
    float* out = (float*)d_out;
    float* ws  = (float*)d_ws;

    float* Qb = ws;
    float* Kb = ws + QSZ;
    float* Vb = ws + 2 * QSZ;
    float* Ab = ws + 3 * QSZ;   // attention output, (B, N, D)

    // 1) fused QKV projection (scatter into split Q/K/V, (B,H,N,hd))
    dim3 g1((3 * DMODEL) / 64, (MROWS + 31) / 32);
    gemm_wmma_kernel<1><<<g1, 256, 0, stream>>>(x, w_qkv, b_qkv, ws, MROWS, 3 * DMODEL, DMODEL);

    // 2) RoPE on q, k (cls token passthrough)
    constexpr int RTOTAL = BATCH * HEADS * (NTOK - 1) * (HD / 2);
    rope_kernel<<<(RTOTAL + 255) / 256, 256, 0, stream>>>(Qb, Kb, position);

    // 3) flash attention
    dim3 g2((NTOK + 127) / 128, BATCH * HEADS);
    attn_kernel<<<g2, 256, 0, stream>>>(Qb, Kb, Vb, Ab, NTOK);

    // 4) output projection
    dim3 g3(DMODEL / 64, (MROWS + 31) / 32);
    gemm_wmma_kernel<0><<<g3, 256, 0, stream>>>(Ab, w_fc, b_fc, out, MROWS, DMODEL, DMODEL);
}